// MinimumSpanningTree_11871289606746
// MI455X (gfx1250) — compile-verified
//
#include <hip/hip_runtime.h>
#include <hip/hip_bf16.h>
#include <stdint.h>

// Problem constants (B, C, H, W) = (8, 64, 256, 256)
#define Hh 256
#define Ww 256
#define Cc 64
#define Bb 8
#define Vv (Hh * Ww)          // 65536 vertices
#define EROW ((Hh - 1) * Ww)  // 65280 row edges
#define Ee (2 * EROW)         // 130560 edges total
#define NROUNDS 16            // ceil(log2 V)
#define NBLK_V (Vv / 256)     // 256
#define NBLK_E ((Ee + 255) / 256) // 510

typedef __attribute__((ext_vector_type(2))) float v2f;
typedef __attribute__((ext_vector_type(8))) float v8f;

__device__ __forceinline__ void edge_uv(int e, int& u, int& v) {
    if (e < EROW) {                 // row edge: (p, p+W)
        u = e; v = e + Ww;
    } else {                        // col edge: (p, p+1)
        int j = e - EROW;
        int r = j / (Ww - 1);
        int c = j - r * (Ww - 1);
        u = r * Ww + c; v = u + 1;
    }
}

// ---------------------------------------------------------------------------
// Edge weights via V_WMMA_F32_16X16X4_F32.
// One wave handles 16 edges. Per K-chunk of 4 channels each lane builds its
// 2-VGPR slice of the 16x4 diff matrix A (lanes 0-15: K=0/1, lanes 16-31:
// K=2/3 per ISA layout). For the symmetric Gram product A*A^T the 4x16 B
// operand register image is identical to A, so we pass A twice. The diagonal
// of the f32 accumulator is exactly sum_c d^2 for each edge.
// ---------------------------------------------------------------------------
__global__ void k_edge_weights_wmma(const float* __restrict__ fm,
                                    float* __restrict__ wgt) {
    const int b    = blockIdx.y;
    const int wave = threadIdx.x >> 5;
    const int lane = threadIdx.x & 31;
    const int e16  = (blockIdx.x * (blockDim.x >> 5) + wave) << 4;
    if (e16 >= Ee) return;                      // wave-uniform, EXEC stays full

    const int m    = lane & 15;                 // edge row of A this lane feeds
    const int koff = (lane >> 4) << 1;          // 0 (lanes 0-15) or 2 (16-31)

    int u, v;
    edge_uv(e16 + m, u, v);
    const float* base = fm + (size_t)b * Cc * Vv;

    v8f acc = {};
#pragma unroll
    for (int c0 = 0; c0 < Cc; c0 += 4) {
        const int c = c0 + koff;
        float d0 = base[(size_t)c * Vv + u]       - base[(size_t)c * Vv + v];
        float d1 = base[(size_t)(c + 1) * Vv + u] - base[(size_t)(c + 1) * Vv + v];
        v2f a; a.x = d0; a.y = d1;
        // D(16x16) += A(16x4) * A^T(4x16); diag(D)[e] accumulates sum d^2
        acc = __builtin_amdgcn_wmma_f32_16x16x4_f32(
            /*neg_a=*/false, a, /*neg_b=*/false, a,
            /*c_mod=*/(short)0, acc, /*reuse_a=*/false, /*reuse_b=*/false);
    }

    // Diagonal extraction per C/D layout: e<8 -> acc[e] @ lane e;
    // e>=8 -> acc[e-8] @ lane e+16.
    int eo, vi;
    if (lane < 8)        { eo = lane;      vi = lane;      }
    else if (lane >= 24) { eo = lane - 16; vi = lane - 24; }
    else return;

    float r;
    switch (vi) {
        case 0: r = acc[0]; break; case 1: r = acc[1]; break;
        case 2: r = acc[2]; break; case 3: r = acc[3]; break;
        case 4: r = acc[4]; break; case 5: r = acc[5]; break;
        case 6: r = acc[6]; break; default: r = acc[7]; break;
    }
    wgt[(size_t)b * Ee + e16 + eo] = r + 1.0f;
}

// ---------------------------------------------------------------------------
// Boruvka kernels (all per-batch via blockIdx.y)
// ---------------------------------------------------------------------------
__global__ void k_init(int* __restrict__ comp, unsigned char* __restrict__ sel,
                       unsigned long long* __restrict__ packed) {
    const int i = blockIdx.x * 256 + threadIdx.x;
    const int b = blockIdx.y;
    if (i < Ee) sel[(size_t)b * Ee + i] = 0;
    if (i < Vv) {
        comp[(size_t)b * Vv + i]   = i;
        packed[(size_t)b * Vv + i] = ~0ull;
    }
}

// Scatter-min of (weight, edge index) to both endpoint components.
// Packed key order == (min weight, then min edge id): matches reference,
// and makes all keys globally distinct -> hooking cycles are exactly 2-cycles.
__global__ void k_edge_min(const int* __restrict__ comp,
                           const float* __restrict__ wgt,
                           unsigned long long* __restrict__ packed) {
    const int e = blockIdx.x * 256 + threadIdx.x;
    if (e >= Ee) return;
    const int b = blockIdx.y;
    const int* cb = comp + (size_t)b * Vv;
    int u, v; edge_uv(e, u, v);
    const int cu = cb[u], cv = cb[v];
    if (cu == cv) return;
    const float w = wgt[(size_t)b * Ee + e];
    const unsigned long long key =
        ((unsigned long long)__float_as_uint(w) << 32) | (unsigned)e;
    unsigned long long* pb = packed + (size_t)b * Vv;
    atomicMin(&pb[cu], key);
    atomicMin(&pb[cv], key);
}

// Hook: parent[v] = other endpoint's component of the winning edge; mark sel.
__global__ void k_select(const unsigned long long* __restrict__ packed,
                         const int* __restrict__ comp,
                         int* __restrict__ parent,
                         unsigned char* __restrict__ sel) {
    const int v = blockIdx.x * 256 + threadIdx.x;
    const int b = blockIdx.y;
    const unsigned long long key = packed[(size_t)b * Vv + v];
    int p = v;
    if (key != ~0ull) {
        const int e = (int)(key & 0xFFFFFFFFu);
        sel[(size_t)b * Ee + e] = 1;            // benign duplicate writes
        int eu, ev; edge_uv(e, eu, ev);
        const int* cb = comp + (size_t)b * Vv;
        const int cu = cb[eu], cv = cb[ev];
        p = (cu == v) ? cv : cu;
    }
    parent[(size_t)b * Vv + v] = p;
}

// Break 2-cycles: keep the smaller id as root (reference semantics).
// After this step the parent graph is a forest (all cycles were 2-cycles).
__global__ void k_break(const int* __restrict__ src, int* __restrict__ dst) {
    const int v = blockIdx.x * 256 + threadIdx.x;
    const size_t o = (size_t)blockIdx.y * Vv;
    const int p  = src[o + v];
    const int pp = src[o + p];
    dst[o + v] = (pp == v && v < p) ? v : p;
}

// comp[v] = root(parent, comp[v]) -- equivalent to the reference's 16 pointer
// doublings (2^16 >= any tree depth) followed by comp = parent[comp].
// Also resets packed[] for the next round (fused per-vertex write).
__global__ void k_comp_root(const int* __restrict__ P, int* __restrict__ comp,
                            unsigned long long* __restrict__ packed) {
    const int v = blockIdx.x * 256 + threadIdx.x;
    const size_t o = (size_t)blockIdx.y * Vv;
    int p = comp[o + v];
    int q = P[o + p];
    int guard = 0;
    while (q != p && guard < Vv) {      // forest: terminates at the root
        p = q;
        q = P[o + p];
        ++guard;
    }
    comp[o + v] = p;
    packed[o + v] = ~0ull;
}

// ---------------------------------------------------------------------------
// Ordered compaction of selected edges -> (B, V-1, 2) int32
// ---------------------------------------------------------------------------
__global__ void k_count(const unsigned char* __restrict__ sel,
                        int* __restrict__ bsum) {
    __shared__ int sh[256];
    const int e = blockIdx.x * 256 + threadIdx.x;
    const int b = blockIdx.y;
    sh[threadIdx.x] = (e < Ee) ? (int)sel[(size_t)b * Ee + e] : 0;
    __syncthreads();
    for (int s = 128; s > 0; s >>= 1) {
        if (threadIdx.x < s) sh[threadIdx.x] += sh[threadIdx.x + s];
        __syncthreads();
    }
    if (threadIdx.x == 0) bsum[b * 1024 + blockIdx.x] = sh[0];
}

__global__ void k_scan(const int* __restrict__ bsum, int* __restrict__ boff) {
    __shared__ int sh[512];
    const int b = blockIdx.x;
    const int t = threadIdx.x;
    const int own = (t < NBLK_E) ? bsum[b * 1024 + t] : 0;
    sh[t] = own;
    __syncthreads();
    for (int d = 1; d < 512; d <<= 1) {
        const int val = (t >= d) ? sh[t - d] : 0;
        __syncthreads();
        sh[t] += val;
        __syncthreads();
    }
    if (t < NBLK_E) boff[b * 1024 + t] = sh[t] - own;   // exclusive prefix
}

__global__ void k_emit(const unsigned char* __restrict__ sel,
                       const int* __restrict__ boff,
                       int* __restrict__ out) {
    __shared__ int sh[256];
    const int e = blockIdx.x * 256 + threadIdx.x;
    const int b = blockIdx.y;
    const int s = (e < Ee) ? (int)sel[(size_t)b * Ee + e] : 0;
    sh[threadIdx.x] = s;
    __syncthreads();
    for (int d = 1; d < 256; d <<= 1) {
        const int val = (threadIdx.x >= d) ? sh[threadIdx.x - d] : 0;
        __syncthreads();
        sh[threadIdx.x] += val;
        __syncthreads();
    }
    if (s) {
        const int pos = boff[b * 1024 + blockIdx.x] + sh[threadIdx.x] - 1;
        if (pos < Vv - 1) {
            int u, v; edge_uv(e, u, v);
            out[((size_t)b * (Vv - 1) + pos) * 2 + 0] = u;
            out[((size_t)b * (Vv - 1) + pos) * 2 + 1] = v;
        }
    }
}

// ---------------------------------------------------------------------------
extern "C" void kernel_launch(void* const* d_in, const int* in_sizes, int n_in,
                              void* d_out, int out_size, void* d_ws, size_t ws_size,
                              hipStream_t stream) {
    (void)in_sizes; (void)n_in; (void)out_size; (void)ws_size;
    const float* fm = (const float*)d_in[0];
    int* out = (int*)d_out;

    // Workspace partition (all 256B aligned)
    char* ws = (char*)d_ws;
    size_t off = 0;
    auto take = [&](size_t bytes) -> void* {
        off = (off + 255) & ~(size_t)255;
        void* p = ws + off;
        off += bytes;
        return p;
    };
    float*              weights = (float*)take((size_t)Bb * Ee * sizeof(float));
    unsigned long long* packed  = (unsigned long long*)take((size_t)Bb * Vv * 8);
    int*                comp    = (int*)take((size_t)Bb * Vv * 4);
    int*                parent  = (int*)take((size_t)Bb * Vv * 4);
    int*                parent2 = (int*)take((size_t)Bb * Vv * 4);
    unsigned char*      sel     = (unsigned char*)take((size_t)Bb * Ee);
    int*                bsum    = (int*)take((size_t)Bb * 1024 * 4);
    int*                boff    = (int*)take((size_t)Bb * 1024 * 4);

    const dim3 blk(256);
    const dim3 gV(NBLK_V, Bb);
    const dim3 gE(NBLK_E, Bb);
    const dim3 gW(Ee / 128, Bb);   // 8 waves/block * 16 edges/wave = 128 edges

    k_init<<<gE, blk, 0, stream>>>(comp, sel, packed);
    k_edge_weights_wmma<<<gW, blk, 0, stream>>>(fm, weights);

    for (int r = 0; r < NROUNDS; ++r) {
        k_edge_min<<<gE, blk, 0, stream>>>(comp, weights, packed);
        k_select<<<gV, blk, 0, stream>>>(packed, comp, parent, sel);
        k_break<<<gV, blk, 0, stream>>>(parent, parent2);
        k_comp_root<<<gV, blk, 0, stream>>>(parent2, comp, packed);
    }

    k_count<<<gE, blk, 0, stream>>>(sel, bsum);
    k_scan<<<dim3(Bb), dim3(512), 0, stream>>>(bsum, boff);
    k_emit<<<gE, blk, 0, stream>>>(sel, boff, out);
}